// TrafficPredictionGNN_33689723469921
// MI455X (gfx1250) — compile-verified
//
#include <hip/hip_runtime.h>
#include <hip/hip_bf16.h>

// Problem constants (reference: N=50000, T=12, IN=2, H=128, OUT=12, E=800000)
#define TN   12
#define IND  2
#define HD   128
#define OUTD 12
#define MR   32      // node rows per workgroup (2 WMMA row tiles)
#define HP   132     // padded LDS row stride (132 mod 64 = 4 -> conflict-free)

typedef float v2f __attribute__((ext_vector_type(2)));
typedef float v8f __attribute__((ext_vector_type(8)));

// fp32 WMMA: D(16x16,f32) = A(16x4,f32) * B(4x16,f32) + C
static __device__ __forceinline__ v8f wmma_f32x4(v2f a, v2f b, v8f c) {
  return __builtin_amdgcn_wmma_f32_16x16x4_f32(
      false, a, false, b, (short)0, c, false, false);
}

// cheap activations: v_exp_f32 + v_rcp_f32 (no libm range-reduction blob)
static __device__ __forceinline__ float sigmoidf_(float x) {
  return __builtin_amdgcn_rcpf(1.0f + __expf(-x));
}
static __device__ __forceinline__ float tanhf_(float x) {
  float xc = fminf(fmaxf(x, -30.f), 30.f);
  float t  = __expf(2.0f * xc);
  return (t - 1.0f) * __builtin_amdgcn_rcpf(t + 1.0f);
}

// ---------------------------------------------------------------------------
// Fused 2-layer LSTM, 32 nodes per workgroup, 8 waves. Wave w owns the
// 16-unit slice [w*16, w*16+16) of each gate (i,f,g,o) -> wave-local cell
// update; cell state c lives in registers, h lives in LDS across all T steps.
// K-loops stay ROLLED (unroll 1) with a 1-step software pipeline on the
// B-fragment loads: bounded register pressure, no spills.
// ---------------------------------------------------------------------------
__global__ __launch_bounds__(256) void lstm_fused(
    const float* __restrict__ x,                                // [N][T][2]
    const float* __restrict__ Wih0, const float* __restrict__ Whh0,
    const float* __restrict__ bih0, const float* __restrict__ bhh0,
    const float* __restrict__ Wih1, const float* __restrict__ Whh1,
    const float* __restrict__ bih1, const float* __restrict__ bhh1,
    float* __restrict__ hout, int N)                            // [N][128]
{
  __shared__ float h0[MR][HP], h1[MR][HP];
  __shared__ float xs[MR][TN][IND];

  const int tid   = threadIdx.x;
  const int lane  = tid & 31;
  const int wave  = tid >> 5;            // 0..7
  const int node0 = blockIdx.x * MR;
  const int colLo = lane & 15;
  const int hi    = lane >> 4;           // lane half
  const int khalf = hi * 2;
  const int rowA  = colLo;

  // stage x (zero-fill tail rows)
  for (int i = tid; i < MR * TN * IND; i += 256) {
    int n = i / (TN * IND);
    float v = 0.f;
    if (node0 + n < N) v = x[(size_t)(node0 + n) * (TN * IND) + (i % (TN * IND))];
    ((float*)xs)[i] = v;
  }
  for (int i = tid; i < MR * HP; i += 256) { ((float*)h0)[i] = 0.f; ((float*)h1)[i] = 0.f; }

  // per-wave gate-column constants
  int   cols[4];
  float bs0[4], bs1[4];
  v2f   wx0[4];
#pragma unroll
  for (int g = 0; g < 4; ++g) {
    int col = g * HD + wave * 16 + colLo;
    cols[g] = col;
    bs0[g]  = bih0[col] + bhh0[col];
    bs1[g]  = bih1[col] + bhh1[col];
    wx0[g]  = *(const v2f*)&Wih0[col * IND];
  }

  // cell state in registers: index [rt*8+v]
  float c0r[16], c1r[16];
#pragma unroll
  for (int i = 0; i < 16; ++i) { c0r[i] = 0.f; c1r[i] = 0.f; }

  __syncthreads();

  const int unit = wave * 16 + colLo;

  for (int t = 0; t < TN; ++t) {
    // ============ layer 0: z = x_t*Wih0^T + h0*Whh0^T + b ============
    v8f acc[4][2];
#pragma unroll
    for (int g = 0; g < 4; ++g)
#pragma unroll
      for (int rt = 0; rt < 2; ++rt) {
        v8f a;
#pragma unroll
        for (int v = 0; v < 8; ++v) {
          int r = rt * 16 + v + (hi << 3);
          a[v] = bs0[g] + xs[r][t][0] * wx0[g].x + xs[r][t][1] * wx0[g].y;
        }
        acc[g][rt] = a;
      }
    {
      v2f bc[4];
#pragma unroll
      for (int g = 0; g < 4; ++g) bc[g] = *(const v2f*)&Whh0[cols[g] * HD + khalf];
#pragma unroll 1
      for (int k0 = 0; k0 < HD; k0 += 4) {
        int kn = (k0 + 4) & (HD - 1);          // wrapped prefetch (stays in bounds)
        v2f bn[4];
#pragma unroll
        for (int g = 0; g < 4; ++g) bn[g] = *(const v2f*)&Whh0[cols[g] * HD + kn + khalf];
        v2f a0 = *(const v2f*)&h0[rowA][k0 + khalf];
        v2f a1 = *(const v2f*)&h0[16 + rowA][k0 + khalf];
#pragma unroll
        for (int g = 0; g < 4; ++g) {
          acc[g][0] = wmma_f32x4(a0, bc[g], acc[g][0]);
          acc[g][1] = wmma_f32x4(a1, bc[g], acc[g][1]);
        }
#pragma unroll
        for (int g = 0; g < 4; ++g) bc[g] = bn[g];
      }
    }
    __syncthreads();   // all h0 reads done before overwriting h0
#pragma unroll
    for (int rt = 0; rt < 2; ++rt)
#pragma unroll
      for (int v = 0; v < 8; ++v) {
        int idx = rt * 8 + v;
        int r   = rt * 16 + v + (hi << 3);
        float cc = sigmoidf_(acc[1][rt][v]) * c0r[idx] +
                   sigmoidf_(acc[0][rt][v]) * tanhf_(acc[2][rt][v]);
        c0r[idx] = cc;
        h0[r][unit] = sigmoidf_(acc[3][rt][v]) * tanhf_(cc);
      }
    __syncthreads();

    // ============ layer 1: z = y0*Wih1^T + h1*Whh1^T + b ============
#pragma unroll
    for (int g = 0; g < 4; ++g)
#pragma unroll
      for (int rt = 0; rt < 2; ++rt) {
        v8f a;
#pragma unroll
        for (int v = 0; v < 8; ++v) a[v] = bs1[g];
        acc[g][rt] = a;
      }
    // --- sweep 1: y0 @ Wih1^T (A from h0) ---
    {
      v2f bc[4];
#pragma unroll
      for (int g = 0; g < 4; ++g) bc[g] = *(const v2f*)&Wih1[cols[g] * HD + khalf];
#pragma unroll 1
      for (int k0 = 0; k0 < HD; k0 += 4) {
        int kn = (k0 + 4) & (HD - 1);
        v2f bn[4];
#pragma unroll
        for (int g = 0; g < 4; ++g) bn[g] = *(const v2f*)&Wih1[cols[g] * HD + kn + khalf];
        v2f a0 = *(const v2f*)&h0[rowA][k0 + khalf];
        v2f a1 = *(const v2f*)&h0[16 + rowA][k0 + khalf];
#pragma unroll
        for (int g = 0; g < 4; ++g) {
          acc[g][0] = wmma_f32x4(a0, bc[g], acc[g][0]);
          acc[g][1] = wmma_f32x4(a1, bc[g], acc[g][1]);
        }
#pragma unroll
        for (int g = 0; g < 4; ++g) bc[g] = bn[g];
      }
    }
    // --- sweep 2: h1 @ Whh1^T (A from h1) ---
    {
      v2f bc[4];
#pragma unroll
      for (int g = 0; g < 4; ++g) bc[g] = *(const v2f*)&Whh1[cols[g] * HD + khalf];
#pragma unroll 1
      for (int k0 = 0; k0 < HD; k0 += 4) {
        int kn = (k0 + 4) & (HD - 1);
        v2f bn[4];
#pragma unroll
        for (int g = 0; g < 4; ++g) bn[g] = *(const v2f*)&Whh1[cols[g] * HD + kn + khalf];
        v2f a0 = *(const v2f*)&h1[rowA][k0 + khalf];
        v2f a1 = *(const v2f*)&h1[16 + rowA][k0 + khalf];
#pragma unroll
        for (int g = 0; g < 4; ++g) {
          acc[g][0] = wmma_f32x4(a0, bc[g], acc[g][0]);
          acc[g][1] = wmma_f32x4(a1, bc[g], acc[g][1]);
        }
#pragma unroll
        for (int g = 0; g < 4; ++g) bc[g] = bn[g];
      }
    }
    __syncthreads();
#pragma unroll
    for (int rt = 0; rt < 2; ++rt)
#pragma unroll
      for (int v = 0; v < 8; ++v) {
        int idx = rt * 8 + v;
        int r   = rt * 16 + v + (hi << 3);
        float cc = sigmoidf_(acc[1][rt][v]) * c1r[idx] +
                   sigmoidf_(acc[0][rt][v]) * tanhf_(acc[2][rt][v]);
        c1r[idx] = cc;
        h1[r][unit] = sigmoidf_(acc[3][rt][v]) * tanhf_(cc);
      }
    __syncthreads();
  }

  // write final layer-1 hidden state (guard tail)
  for (int i = tid; i < MR * HD; i += 256) {
    int n = i >> 7, u = i & 127;
    if (node0 + n < N) hout[(size_t)(node0 + n) * HD + u] = h1[n][u];
  }
}

// ---------------------------------------------------------------------------
// GCN: m = (h @ W) * dinv — 32 rows per block, pipelined B, 2 WMMAs per B frag
// ---------------------------------------------------------------------------
__global__ void gcn_matmul(const float* __restrict__ h,   // [N][128]
                           const float* __restrict__ W,   // [128][C]
                           const float* __restrict__ dinv,
                           float* __restrict__ m,         // [N][C]
                           int C, int N)
{
  const int lane  = threadIdx.x & 31;
  const int wave  = threadIdx.x >> 5;
  const int node0 = blockIdx.x * MR;
  const int colLo = lane & 15;
  const int hi    = lane >> 4;
  const int khalf = hi * 2;
  const int rowA  = colLo;
  const int col   = wave * 16 + colLo;

  v8f acc0 = {0.f,0.f,0.f,0.f,0.f,0.f,0.f,0.f};
  v8f acc1 = acc0;
  v2f bc;
  bc.x = (col < C) ? W[khalf * C + col]       : 0.f;
  bc.y = (col < C) ? W[(khalf + 1) * C + col] : 0.f;
#pragma unroll 1
  for (int k0 = 0; k0 < HD; k0 += 4) {
    int kn = (k0 + 4) & (HD - 1);
    v2f bn;
    bn.x = (col < C) ? W[(kn + khalf) * C + col]     : 0.f;
    bn.y = (col < C) ? W[(kn + khalf + 1) * C + col] : 0.f;
    // tail rows read past h[N] but stay inside d_ws; stores are guarded
    v2f a0 = *(const v2f*)&h[(size_t)(node0 + rowA) * HD + k0 + khalf];
    v2f a1 = *(const v2f*)&h[(size_t)(node0 + 16 + rowA) * HD + k0 + khalf];
    acc0 = wmma_f32x4(a0, bc, acc0);
    acc1 = wmma_f32x4(a1, bc, acc1);
    bc = bn;
  }
#pragma unroll
  for (int v = 0; v < 8; ++v) {
    int r0 = v + (hi << 3), r1 = 16 + r0;
    if (col < C && node0 + r0 < N)
      m[(size_t)(node0 + r0) * C + col] = acc0[v] * dinv[node0 + r0];
    if (col < C && node0 + r1 < N)
      m[(size_t)(node0 + r1) * C + col] = acc1[v] * dinv[node0 + r1];
  }
}

// one wave per edge; lanes cover the C features (coalesced)
__global__ void gcn_scatter(const float* __restrict__ m, const int* __restrict__ ei,
                            float* __restrict__ acc, int E, int C)
{
  int e = blockIdx.x * 8 + (threadIdx.x >> 5);
  if (e >= E) return;
  int lane = threadIdx.x & 31;
  int src = ei[e], dst = ei[E + e];
  const float* ms = m + (size_t)src * C;
  float* ad = acc + (size_t)dst * C;
  for (int j = lane; j < C; j += 32)
    atomicAdd(&ad[j], ms[j]);
}

// out = dinv*(acc + m_self) + b, optional relu
__global__ void gcn_finish(const float* __restrict__ accv, const float* __restrict__ m,
                           const float* __restrict__ dinv, const float* __restrict__ bias,
                           float* __restrict__ out, int total, int C, int relu)
{
  int i = blockIdx.x * blockDim.x + threadIdx.x;
  if (i >= total) return;
  int n = i / C, j = i - n * C;
  float v = dinv[n] * (accv[i] + m[i]) + bias[j];
  out[i] = relu ? fmaxf(v, 0.f) : v;
}

__global__ void zero_f32(float* __restrict__ p, int n) {
  int i = blockIdx.x * blockDim.x + threadIdx.x;
  if (i < n) p[i] = 0.f;
}

__global__ void deg_scatter(const int* __restrict__ ei, float* __restrict__ deg, int E) {
  int e = blockIdx.x * blockDim.x + threadIdx.x;
  if (e < E) atomicAdd(&deg[ei[E + e]], 1.0f);   // exact integer-valued adds
}

__global__ void compute_dinv(const float* __restrict__ deg, float* __restrict__ dinv, int n) {
  int i = blockIdx.x * blockDim.x + threadIdx.x;
  if (i < n) dinv[i] = rsqrtf(deg[i] + 1.0f);    // +1 self loop
}

// ---------------------------------------------------------------------------
extern "C" void kernel_launch(void* const* d_in, const int* in_sizes, int n_in,
                              void* d_out, int out_size, void* d_ws, size_t ws_size,
                              hipStream_t stream) {
  const float* x    = (const float*)d_in[0];
  const int*   ei   = (const int*)  d_in[1];
  const float* Wih0 = (const float*)d_in[2];
  const float* Whh0 = (const float*)d_in[3];
  const float* bih0 = (const float*)d_in[4];
  const float* bhh0 = (const float*)d_in[5];
  const float* Wih1 = (const float*)d_in[6];
  const float* Whh1 = (const float*)d_in[7];
  const float* bih1 = (const float*)d_in[8];
  const float* bhh1 = (const float*)d_in[9];
  const float* W0   = (const float*)d_in[10];
  const float* b0   = (const float*)d_in[11];
  const float* W1   = (const float*)d_in[12];
  const float* b1   = (const float*)d_in[13];
  const float* W2   = (const float*)d_in[14];
  const float* b2   = (const float*)d_in[15];

  const int N = in_sizes[0] / (TN * IND);   // 50000
  const int E = in_sizes[1] / 2;            // 800000
  const int nTiles = (N + MR - 1) / MR;

  // workspace layout (~77 MB)
  float* hbuf = (float*)d_ws;                 // N*128
  float* mbuf = hbuf + (size_t)N * HD;        // N*128
  float* abuf = mbuf + (size_t)N * HD;        // N*128
  float* dinv = abuf + (size_t)N * HD;        // N
  float* deg  = dinv + N;                     // N

  // degree / normalization
  zero_f32    <<<(N + 255) / 256, 256, 0, stream>>>(deg, N);
  deg_scatter <<<(E + 255) / 256, 256, 0, stream>>>(ei, deg, E);
  compute_dinv<<<(N + 255) / 256, 256, 0, stream>>>(deg, dinv, N);

  // fused 2-layer LSTM -> hbuf
  lstm_fused<<<nTiles, 256, 0, stream>>>(x, Wih0, Whh0, bih0, bhh0,
                                         Wih1, Whh1, bih1, bhh1, hbuf, N);

  // GCN layers
  struct Layer { const float* W; const float* b; int C; int relu; float* out; };
  Layer L[3] = { {W0, b0, HD,   1, hbuf},
                 {W1, b1, HD,   1, hbuf},
                 {W2, b2, OUTD, 0, (float*)d_out} };
  for (int l = 0; l < 3; ++l) {
    int C = L[l].C;
    int tiles = (C + 15) / 16;
    int tot = N * C;
    gcn_matmul <<<nTiles, 32 * tiles, 0, stream>>>(hbuf, L[l].W, dinv, mbuf, C, N);
    zero_f32   <<<(tot + 255) / 256, 256, 0, stream>>>(abuf, tot);
    gcn_scatter<<<(E + 7) / 8, 256, 0, stream>>>(mbuf, ei, abuf, E, C);
    gcn_finish <<<(tot + 255) / 256, 256, 0, stream>>>(abuf, mbuf, dinv, L[l].b,
                                                       L[l].out, tot, C, L[l].relu);
  }
}